// SpaceFormer_49134425866841
// MI455X (gfx1250) — compile-verified
//
#include <hip/hip_runtime.h>
#include <hip/hip_bf16.h>
#include <math.h>

// ---------------------------------------------------------------------------
// SpaceFormer forward for MI455X (gfx1250, wave32, WMMA f32 16x16x4).
// N=2048 cells, D=2000 genes, DL=DG=32. Sparse local attention (~20 nbrs/row)
// computed as a gather instead of the reference's dense [N,32,N] cube.
// LDS layouts tuned: A rows padded to 20 dwords (bank-conflict-free, 16B
// aligned stores); B stored K-pair interleaved with 162-dword pair stride so
// each WMMA B fragment is one aligned ds_load_b64 (no v_mov repacking).
// ---------------------------------------------------------------------------

#define NN   2048
#define DD   2000
#define DLG  32
#define KEXT 2064   // D + DL + DG, divisible by 16
#define LDW  2064   // leading dim of W_ext and P_ext

#define ASTRIDE 20   // A tile row stride in dwords (128 x 20 = 2560 floats)
#define BSTRIDE 162  // B tile k-pair stride in dwords (8 x 162 = 1296 floats)

typedef __attribute__((ext_vector_type(2))) float v2f;
typedef __attribute__((ext_vector_type(8))) float v8f;

__device__ __forceinline__ float nonneg_f(float v) {
    // elu(v) + 1
    return v > 0.0f ? v + 1.0f : expf(v);
}

// --- W_ext[j][k] = nonneg( [vego_w | vl_w | vg_w] ), j<2000 rows, k<2064 cols.
// The 2000x2000 block (ld=2064) is W_ego row-major, reused by both GEMMs.
__global__ void build_wext_kernel(const float* __restrict__ vego,
                                  const float* __restrict__ vlw,
                                  const float* __restrict__ vgw,
                                  float* __restrict__ Wext) {
    const int j = blockIdx.x;                       // 0..1999
    for (int k = threadIdx.x; k < KEXT; k += 256) {
        float v;
        if (k < DD)            v = vego[(size_t)j * DD + k];
        else if (k < DD + DLG) v = vlw[j * DLG + (k - DD)];
        else                   v = vgw[j * DLG + (k - DD - DLG)];
        Wext[(size_t)j * LDW + k] = nonneg_f(v);
    }
}

// --- x_bar[c] = mean over rows of x
__global__ void colmean_kernel(const float* __restrict__ x, float* __restrict__ xbar) {
    const int c = blockIdx.x * blockDim.x + threadIdx.x;
    if (c < DD) {
        float s = 0.0f;
        for (int i = 0; i < NN; ++i) s += x[(size_t)i * DD + c];
        xbar[c] = s * (1.0f / (float)NN);
    }
}

// --- rowsum[i] = sum over D of masked_x row (for the +shift term)
__global__ void rowsum_kernel(const float* __restrict__ mx, float* __restrict__ rs) {
    __shared__ float red[256];
    const int i = blockIdx.x, t = threadIdx.x;
    float s = 0.0f;
    for (int j = t; j < DD; j += 256) s += mx[(size_t)i * DD + j];
    red[t] = s; __syncthreads();
    for (int st = 128; st > 0; st >>= 1) {
        if (t < st) red[t] += red[t + st];
        __syncthreads();
    }
    if (t == 0) rs[i] = red[0];
}

// --- q_l = mx@ql_w.T ; k_l = x@nonneg(kl_w).T ; q_g = mx@qg_w.T  (all [N,32])
__global__ void proj_kernel(const float* __restrict__ x, const float* __restrict__ mx,
                            const float* __restrict__ qlw, const float* __restrict__ klw,
                            const float* __restrict__ qgw,
                            float* __restrict__ ql, float* __restrict__ kl,
                            float* __restrict__ qg) {
    const int which = blockIdx.x;       // 0:q_l 1:k_l 2:q_g
    const int i = blockIdx.y;
    const int c = threadIdx.x;          // 0..31 (one wave32)
    const float* xin = (which == 1) ? x : mx;
    const float* wp  = (which == 0) ? qlw : ((which == 1) ? klw : qgw);
    const bool apply_nn = (which == 1);
    float s = 0.0f;
    const float* xrow = xin + (size_t)i * DD;
    const float* wrow = wp + (size_t)c * DD;
    for (int d = 0; d < DD; ++d) {
        float w = wrow[d];
        if (apply_nn) w = nonneg_f(w);
        s += xrow[d] * w;               // xrow[d] broadcasts across lanes
    }
    float* outp = (which == 0) ? ql : ((which == 1) ? kl : qg);
    outp[i * DLG + c] = s;
}

// --- k_g[c] = x_bar . nonneg(kg_w[c,:])
__global__ void kg_kernel(const float* __restrict__ xbar, const float* __restrict__ kgw,
                          float* __restrict__ kg) {
    const int c = threadIdx.x;          // 32 threads
    float s = 0.0f;
    const float* wrow = kgw + (size_t)c * DD;
    for (int d = 0; d < DD; ++d) s += xbar[d] * nonneg_f(wrow[d]);
    kg[c] = s;
}

// ---------------------------------------------------------------------------
// GEMM1: P_ext[:, :2000] = (masked_x @ W + shift*rowsum) / D
// Tiles: WG = 128x80, wave = 16x80 (5 accumulators), K staged 16 via LDS.
// WMMA f32 16x16x4 layouts per ISA 7.12.2:
//   A: lane(l&15)=row; VGPR0 holds K=kk (+2 for upper lanes), VGPR1 K=kk+1.
//   C/D: VGPR r -> row r (lanes 0-15) / r+8 (lanes 16-31), col = l&15.
// ---------------------------------------------------------------------------
__global__ __launch_bounds__(256)
void gemm1_kernel(const float* __restrict__ X, const float* __restrict__ Wext,
                  const float* __restrict__ rowsum, const float* __restrict__ shiftp,
                  float* __restrict__ Pext) {
    __shared__ float As[128 * ASTRIDE];
    __shared__ float Bs[8 * BSTRIDE];
    const int t    = threadIdx.x;
    const int wv   = t >> 5;
    const int lane = t & 31;
    const int l15  = lane & 15;
    const int half = lane >> 4;
    const int rowBase = blockIdx.y * 128;
    const int colBase = blockIdx.x * 80;

    v8f zero = {};
    v8f acc[5];
    for (int n = 0; n < 5; ++n) acc[n] = zero;

    const int ar = t >> 1, acb = (t & 1) * 8;
    const int arowOff = (wv * 16 + l15) * ASTRIDE;
    for (int kb = 0; kb < DD; kb += 16) {
        // Stage A 128x16 (float4 global+LDS, 16B aligned; ASTRIDE=20 -> no bank conflicts)
        {
            const float4* src = reinterpret_cast<const float4*>(
                &X[(size_t)(rowBase + ar) * DD + kb + acb]);
            float4* dst = reinterpret_cast<float4*>(&As[ar * ASTRIDE + acb]);
            dst[0] = src[0];
            dst[1] = src[1];
        }
        // Stage B 16x80, K-pair interleaved: Bs[(r/2)*162 + c*2 + (r&1)]
        for (int idx = t; idx < 1280; idx += 256) {
            int r = idx / 80, c = idx - r * 80;
            Bs[(r >> 1) * BSTRIDE + c * 2 + (r & 1)] =
                Wext[(size_t)(kb + r) * LDW + colBase + c];
        }
        __syncthreads();
        for (int k4 = 0; k4 < 4; ++k4) {
            const int kk = k4 * 4 + half * 2;
            v2f a = *reinterpret_cast<const v2f*>(&As[arowOff + kk]);
            const int bbase = (k4 * 2 + half) * BSTRIDE + l15 * 2;
            for (int n = 0; n < 5; ++n) {
                v2f b = *reinterpret_cast<const v2f*>(&Bs[bbase + n * 32]);
                acc[n] = __builtin_amdgcn_wmma_f32_16x16x4_f32(
                    false, a, false, b, (short)0, acc[n], false, false);
            }
        }
        __syncthreads();
    }
    const float shift = shiftp[0];
    const float scale = 1.0f / (float)DD;
    for (int n = 0; n < 5; ++n) {
        const int gcol = colBase + n * 16 + l15;
        for (int r = 0; r < 8; ++r) {
            const int grow = rowBase + wv * 16 + (half ? (r + 8) : r);
            Pext[(size_t)grow * LDW + gcol] = (acc[n][r] + shift * rowsum[grow]) * scale;
        }
    }
}

// ---------------------------------------------------------------------------
// Fused per-row softmax: max over {ego row, sparse local scores, global scores},
// then exp in place into P_ext row (ego cols 0..1999, local_s 2000..2031,
// g_s 2032..2063) and the total sum. One workgroup per row.
// ---------------------------------------------------------------------------
__global__ __launch_bounds__(256)
void softmax_kernel(const unsigned char* __restrict__ adj,
                    const float* __restrict__ ql, const float* __restrict__ kl,
                    const float* __restrict__ qg, const float* __restrict__ kg,
                    float* __restrict__ Pext, float* __restrict__ sum_score) {
    __shared__ float red[256];
    __shared__ float lsacc[256];
    __shared__ float qlrow[32];
    __shared__ float msh;
    const int i = blockIdx.x, t = threadIdx.x;
    const float inv_d2 = 1.0f / ((float)DD * (float)DD);
    float* Prow = Pext + (size_t)i * LDW;
    const unsigned char* arow = adj + (size_t)i * NN;

    if (t < 32) qlrow[t] = ql[i * DLG + t];
    __syncthreads();

    // ---- phase A: row max (init -1e9 matches the reference's mask value) ----
    float m = -1e9f;
    for (int j = t; j < DD; j += 256) m = fmaxf(m, Prow[j]);
    for (int j = t; j < NN; j += 256) {
        if (arow[j]) {
            const float* klr = kl + j * DLG;
            for (int c = 0; c < DLG; ++c) m = fmaxf(m, qlrow[c] * klr[c] * inv_d2);
        }
    }
    float gsc = 0.0f;
    if (t < 32) { gsc = qg[i * DLG + t] * kg[t] * inv_d2; m = fmaxf(m, gsc); }
    red[t] = m; __syncthreads();
    for (int st = 128; st > 0; st >>= 1) {
        if (t < st) red[t] = fmaxf(red[t], red[t + st]);
        __syncthreads();
    }
    if (t == 0) msh = red[0];
    __syncthreads();
    const float ms = msh;

    // ---- phase B: exp + sums ----
    float partial = 0.0f;
    for (int j = t; j < DD; j += 256) {
        float e = expf(Prow[j] - ms);
        Prow[j] = e;
        partial += e;
    }
    const int c = t & 31, grp = t >> 5;            // 8 j-groups x 32 channels
    float lacc = 0.0f;
    for (int j = grp; j < NN; j += 8) {
        if (arow[j]) lacc += expf(qlrow[c] * kl[j * DLG + c] * inv_d2 - ms);
    }
    lsacc[t] = lacc; __syncthreads();
    if (t < 32) {
        float ls = 0.0f;
        for (int g = 0; g < 8; ++g) ls += lsacc[g * 32 + t];
        Prow[DD + t] = ls;                          // local_s
    }
    float gv = 0.0f;
    if (t < 32) { gv = expf(gsc - ms); Prow[DD + DLG + t] = gv; }   // g_s
    red[t] = partial + lacc + gv; __syncthreads();
    for (int st = 128; st > 0; st >>= 1) {
        if (t < st) red[t] += red[t + st];
        __syncthreads();
    }
    if (t == 0) sum_score[i] = red[0];
}

// ---------------------------------------------------------------------------
// GEMM2: out[i][j] = (sum_k P_ext[i,k] * Wext[j,k]) / sum_score[i] + nonneg(bias[j])
// K = 2064 fuses the ego/local/global heads. B staged transposed (r-fastest
// global reads stay contiguous); K-pair interleaved LDS layout is write
// bank-conflict-free under the 162-dword pair stride.
// ---------------------------------------------------------------------------
__global__ __launch_bounds__(256)
void gemm2_kernel(const float* __restrict__ Pext, const float* __restrict__ Wext,
                  const float* __restrict__ sum_score, const float* __restrict__ biasb,
                  float* __restrict__ out) {
    __shared__ float As[128 * ASTRIDE];
    __shared__ float Bs[8 * BSTRIDE];
    const int t    = threadIdx.x;
    const int wv   = t >> 5;
    const int lane = t & 31;
    const int l15  = lane & 15;
    const int half = lane >> 4;
    const int rowBase = blockIdx.y * 128;
    const int colBase = blockIdx.x * 80;

    v8f zero = {};
    v8f acc[5];
    for (int n = 0; n < 5; ++n) acc[n] = zero;

    const int ar = t >> 1, acb = (t & 1) * 8;
    const int arowOff = (wv * 16 + l15) * ASTRIDE;
    for (int kb = 0; kb < KEXT; kb += 16) {
        {
            const float4* src = reinterpret_cast<const float4*>(
                &Pext[(size_t)(rowBase + ar) * LDW + kb + acb]);
            float4* dst = reinterpret_cast<float4*>(&As[ar * ASTRIDE + acb]);
            dst[0] = src[0];
            dst[1] = src[1];
        }
        // Transposed stage: Bs[(r/2)*162 + c*2 + (r&1)] = Wext[colBase+c][kb+r]
        for (int idx = t; idx < 1280; idx += 256) {
            int r = idx & 15, c = idx >> 4;        // r fastest -> coalesced global
            Bs[(r >> 1) * BSTRIDE + c * 2 + (r & 1)] =
                Wext[(size_t)(colBase + c) * LDW + kb + r];
        }
        __syncthreads();
        for (int k4 = 0; k4 < 4; ++k4) {
            const int kk = k4 * 4 + half * 2;
            v2f a = *reinterpret_cast<const v2f*>(&As[arowOff + kk]);
            const int bbase = (k4 * 2 + half) * BSTRIDE + l15 * 2;
            for (int n = 0; n < 5; ++n) {
                v2f b = *reinterpret_cast<const v2f*>(&Bs[bbase + n * 32]);
                acc[n] = __builtin_amdgcn_wmma_f32_16x16x4_f32(
                    false, a, false, b, (short)0, acc[n], false, false);
            }
        }
        __syncthreads();
    }
    for (int n = 0; n < 5; ++n) {
        const int gcol = colBase + n * 16 + l15;
        const float bias = nonneg_f(biasb[gcol]);
        for (int r = 0; r < 8; ++r) {
            const int grow = rowBase + wv * 16 + (half ? (r + 8) : r);
            out[(size_t)grow * DD + gcol] = acc[n][r] / sum_score[grow] + bias;
        }
    }
}

// ---------------------------------------------------------------------------
extern "C" void kernel_launch(void* const* d_in, const int* in_sizes, int n_in,
                              void* d_out, int out_size, void* d_ws, size_t ws_size,
                              hipStream_t stream) {
    (void)in_sizes; (void)n_in; (void)out_size; (void)ws_size;
    const unsigned char* adj = (const unsigned char*)d_in[0];   // bool [N,N]
    const float* x     = (const float*)d_in[1];
    const float* mx    = (const float*)d_in[2];
    const float* vego  = (const float*)d_in[3];
    const float* shift = (const float*)d_in[4];
    const float* qlw   = (const float*)d_in[5];
    const float* klw   = (const float*)d_in[6];
    const float* vlw   = (const float*)d_in[7];
    const float* qgw   = (const float*)d_in[8];
    const float* kgw   = (const float*)d_in[9];
    const float* vgw   = (const float*)d_in[10];
    const float* biasb = (const float*)d_in[11];
    float* out = (float*)d_out;

    // Workspace carve-up (all regions 16B aligned; ~34.3 MB total)
    float* ws   = (float*)d_ws;
    float* Wext = ws;                                   // 2000*2064
    float* Pext = Wext + (size_t)DD * LDW;              // 2048*2064
    float* ql   = Pext + (size_t)NN * LDW;              // 2048*32
    float* kl   = ql + NN * DLG;
    float* qg   = kl + NN * DLG;
    float* kg   = qg + NN * DLG;                        // 32
    float* xbar = kg + DLG;                             // 2000
    float* rs   = xbar + DD;                            // 2048
    float* sums = rs + NN;                              // 2048

    build_wext_kernel<<<DD, 256, 0, stream>>>(vego, vlw, vgw, Wext);
    colmean_kernel<<<(DD + 255) / 256, 256, 0, stream>>>(x, xbar);
    rowsum_kernel<<<NN, 256, 0, stream>>>(mx, rs);
    proj_kernel<<<dim3(3, NN), 32, 0, stream>>>(x, mx, qlw, klw, qgw, ql, kl, qg);
    kg_kernel<<<1, 32, 0, stream>>>(xbar, kgw, kg);
    gemm1_kernel<<<dim3(DD / 80, NN / 128), 256, 0, stream>>>(mx, Wext, rs, shift, Pext);
    softmax_kernel<<<NN, 256, 0, stream>>>(adj, ql, kl, qg, kg, Pext, sums);
    gemm2_kernel<<<dim3(DD / 80, NN / 128), 256, 0, stream>>>(Pext, Wext, sums, biasb, out);
}